// ConvSpikingBlock_15264313770111
// MI455X (gfx1250) — compile-verified
//
#include <hip/hip_runtime.h>

typedef __attribute__((ext_vector_type(2))) float v2f;
typedef __attribute__((ext_vector_type(8))) float v8f;

#define BB   16
#define TT   20
#define CIN  2
#define COUT 32
#define HH   64
#define WW   64
#define KK   18                       // Cin*3*3 im2col depth (padded to 20 for K-chunks of 4)
#define NPIX (BB * TT * HH * WW)      // per-channel reduction count for tdBN

// ------------------------------------------------------------------
// Pass 1: conv (bias-free; conv bias cancels inside tdBN) + per-channel
// sum / sum-of-squares. One wave per pixel, lane = output channel, so the
// 18 x-loads per pixel are wave-uniform broadcasts.
// ------------------------------------------------------------------
__global__ __launch_bounds__(256) void snn_stats_kernel(
    const float* __restrict__ x, const float* __restrict__ cw,
    float* __restrict__ gsum, float* __restrict__ gsq)
{
  const int tid = threadIdx.x;
  const int c   = tid & 31;

  float wr[KK];
#pragma unroll
  for (int k = 0; k < KK; ++k) wr[k] = cw[c * KK + k];

  const int wave   = blockIdx.x * (blockDim.x >> 5) + (tid >> 5);
  const int nwaves = gridDim.x * (blockDim.x >> 5);

  float s = 0.f, sq = 0.f;
  for (int p = wave; p < NPIX; p += nwaves) {
    const int wpos = p & 63;
    const int hpos = (p >> 6) & 63;
    const int f    = p >> 12;            // global frame index (b*T + t)
    const float* fr = x + (size_t)(f * CIN) * (HH * WW);
    float y = 0.f;
#pragma unroll
    for (int k = 0; k < KK; ++k) {
      const int ci = k / 9, r = k % 9, kh = r / 3, kw = r % 3;
      const int hs = hpos + kh - 1, ws = wpos + kw - 1;
      float v = 0.f;
      if (hs >= 0 && hs < HH && ws >= 0 && ws < WW)
        v = fr[ci * (HH * WW) + hs * WW + ws];
      y = fmaf(wr[k], v, y);
    }
    s += y;
    sq = fmaf(y, y, sq);
  }

  __shared__ float ls[256], lq[256];
  ls[tid] = s; lq[tid] = sq;
  __syncthreads();
  if (tid < 32) {
    float ts = 0.f, tq = 0.f;
#pragma unroll
    for (int i = 0; i < 8; ++i) { ts += ls[tid + 32 * i]; tq += lq[tid + 32 * i]; }
    atomicAdd(&gsum[tid], ts);
    atomicAdd(&gsq[tid], tq);
  }
}

// ------------------------------------------------------------------
// Pass 2: fold tdBN into per-channel scale/shift.
// normalized = (y0 - mean)*rsqrt(var+eps)*gamma + bn_bias  (ALPHA*VTH = 1)
// ------------------------------------------------------------------
__global__ void snn_finalize_kernel(
    const float* __restrict__ gsum, const float* __restrict__ gsq,
    const float* __restrict__ gamma, const float* __restrict__ bnb,
    float* __restrict__ scale, float* __restrict__ shift)
{
  const int c = threadIdx.x;
  if (c < COUT) {
    const float invN = 1.f / (float)NPIX;
    const float mean = gsum[c] * invN;
    const float var  = fmaxf(gsq[c] * invN - mean * mean, 0.f);
    const float inv  = rsqrtf(var + 1e-5f);
    const float sc   = gamma[c] * inv;
    scale[c] = sc;
    shift[c] = bnb[c] - mean * sc;
  }
}

// ------------------------------------------------------------------
// Pass 3: fused conv(WMMA f32 16x16x4) + tdBN + LIF scan over T.
// One wave owns a 16-pixel row tile x all 32 channels for one batch image.
// GEMM: D[ch=16, pix=16] += A(weights 16x4) x B(im2col 4x16), 5 K-chunks,
// two channel tiles. D lane axis = pixel -> coalesced spike stores.
// Membrane state stays in 16 VGPRs across the 20-step temporal loop.
// ------------------------------------------------------------------
__global__ __launch_bounds__(256) void snn_lif_wmma_kernel(
    const float* __restrict__ x, const float* __restrict__ mem0,
    const float* __restrict__ cw, const float* __restrict__ scale,
    const float* __restrict__ shift, const float* __restrict__ pbeta,
    const float* __restrict__ pth, float* __restrict__ out)
{
  const int lane = threadIdx.x & 31;
  const int half = lane >> 4;           // 0: lanes 0-15, 1: lanes 16-31
  const int lm   = lane & 15;
  const int wave = blockIdx.x * 8 + (threadIdx.x >> 5);   // 4096 waves total
  const int wt   = wave & 3;            // 16-pixel tile along W
  const int h    = (wave >> 2) & 63;
  const int b    = wave >> 8;
  const int wbase = wt * 16 + lm;       // this lane's pixel column

  const float beta = fminf(fmaxf(pbeta[0], 0.f), 1.f);
  const float th   = pth[0];

  // A tiles (weights): M=channel, K chunks of 4 (K=18 padded to 20).
  // Lane layout: lanes 0-15 hold K=4kc+{0,1}, lanes 16-31 hold K=4kc+{2,3}.
  v2f a0[5], a1[5];
#pragma unroll
  for (int kc = 0; kc < 5; ++kc) {
    const int kb = 4 * kc + 2 * half;
    v2f t0, t1;
    t0.x = (kb     < KK) ? cw[lm * KK + kb]            : 0.f;
    t0.y = (kb + 1 < KK) ? cw[lm * KK + kb + 1]        : 0.f;
    t1.x = (kb     < KK) ? cw[(16 + lm) * KK + kb]     : 0.f;
    t1.y = (kb + 1 < KK) ? cw[(16 + lm) * KK + kb + 1] : 0.f;
    a0[kc] = t0; a1[kc] = t1;
  }

  // im2col offsets + validity for this lane's pixel (t-invariant, hoisted).
  int off[10]; bool ok[10];
#pragma unroll
  for (int kc = 0; kc < 5; ++kc) {
#pragma unroll
    for (int j = 0; j < 2; ++j) {
      const int k = 4 * kc + 2 * half + j;
      bool v = false; int o = 0;
      if (k < KK) {
        const int ci = k / 9, r = k % 9, kh = r / 3, kw = r % 3;
        const int hs = h + kh - 1, wsrc = wbase + kw - 1;
        v = (hs >= 0) & (hs < HH) & (wsrc >= 0) & (wsrc < WW);
        o = ci * (HH * WW) + hs * WW + wsrc;
      }
      off[2 * kc + j] = o; ok[2 * kc + j] = v;
    }
  }

  // D layout: VGPR r -> (channel = cb + r, pixel = lane&15), cb depends on half.
  const int cb0 = 8 * half;             // channel tile 0: channels cb0 + r
  const int cb1 = 16 + 8 * half;        // channel tile 1
  const size_t pixoff = (size_t)h * WW + wbase;

  float sc0[8], sh0[8], sc1[8], sh1[8], m0[8], m1[8];
#pragma unroll
  for (int r = 0; r < 8; ++r) {
    sc0[r] = scale[cb0 + r]; sh0[r] = shift[cb0 + r];
    sc1[r] = scale[cb1 + r]; sh1[r] = shift[cb1 + r];
    m0[r] = mem0[(size_t)(b * COUT + cb0 + r) * (HH * WW) + pixoff];
    m1[r] = mem0[(size_t)(b * COUT + cb1 + r) * (HH * WW) + pixoff];
  }

  for (int t = 0; t < TT; ++t) {
    const float* fr = x + (size_t)((b * TT + t) * CIN) * (HH * WW);
    v8f acc0 = {}; v8f acc1 = {};
#pragma unroll
    for (int kc = 0; kc < 5; ++kc) {
      v2f bb;
      bb.x = ok[2 * kc]     ? fr[off[2 * kc]]     : 0.f;
      bb.y = ok[2 * kc + 1] ? fr[off[2 * kc + 1]] : 0.f;
      acc0 = __builtin_amdgcn_wmma_f32_16x16x4_f32(false, a0[kc], false, bb,
                                                   (short)0, acc0, false, false);
      acc1 = __builtin_amdgcn_wmma_f32_16x16x4_f32(false, a1[kc], false, bb,
                                                   (short)0, acc1, false, false);
    }

    float* op = out + (size_t)(b * TT + t) * COUT * (HH * WW) + pixoff;
#pragma unroll
    for (int r = 0; r < 8; ++r) {
      // y = conv*scale + shift ; reset = spike(mem - th) ;
      // mem = beta*mem + y - reset*th ; spk = spike(mem - th)
      float y     = fmaf(acc0[r], sc0[r], sh0[r]);
      float reset = (m0[r] > th) ? th : 0.f;
      m0[r] = fmaf(beta, m0[r], y) - reset;
      op[(size_t)(cb0 + r) * (HH * WW)] = (m0[r] > th) ? 1.f : 0.f;

      y     = fmaf(acc1[r], sc1[r], sh1[r]);
      reset = (m1[r] > th) ? th : 0.f;
      m1[r] = fmaf(beta, m1[r], y) - reset;
      op[(size_t)(cb1 + r) * (HH * WW)] = (m1[r] > th) ? 1.f : 0.f;
    }
  }
}

// ------------------------------------------------------------------
extern "C" void kernel_launch(void* const* d_in, const int* in_sizes, int n_in,
                              void* d_out, int out_size, void* d_ws, size_t ws_size,
                              hipStream_t stream) {
  (void)in_sizes; (void)n_in; (void)out_size; (void)ws_size;
  const float* x     = (const float*)d_in[0];
  const float* mem0  = (const float*)d_in[1];
  const float* cw    = (const float*)d_in[2];
  // d_in[3] = conv bias: a per-channel constant, cancelled exactly by tdBN's
  // per-channel mean subtraction -> intentionally unused.
  const float* gamma = (const float*)d_in[4];
  const float* bnb   = (const float*)d_in[5];
  const float* beta  = (const float*)d_in[6];
  const float* th    = (const float*)d_in[7];
  float* out = (float*)d_out;

  float* ws    = (float*)d_ws;
  float* gsum  = ws;        // [32]
  float* gsq   = ws + 32;   // [32]
  float* scale = ws + 64;   // [32]
  float* shift = ws + 96;   // [32]

  hipMemsetAsync(d_ws, 0, 64 * sizeof(float), stream);  // zero accumulators
  snn_stats_kernel<<<1024, 256, 0, stream>>>(x, cw, gsum, gsq);
  snn_finalize_kernel<<<1, 64, 0, stream>>>(gsum, gsq, gamma, bnb, scale, shift);
  // exactly 16(b) * 64(h) * 4(wtile) = 4096 waves -> full EXEC for WMMA
  snn_lif_wmma_kernel<<<512, 256, 0, stream>>>(x, mem0, cw, scale, shift,
                                               beta, th, out);
}